// AddEdgeIndex_35502199669319
// MI455X (gfx1250) — compile-verified
//
#include <hip/hip_runtime.h>
#include <math.h>

// kNN edge construction, fused: WMMA f32 16x16x4 Gram tiles + in-register top-32.
// B graphs x M=1024 nodes x 3D. Grid: (M/16, B), block: 512 threads (16 wave32s).
// Coordinates staged in LDS as float4(x,y,z,|p|^2) -> branch-free WMMA operand
// setup (pure v_cndmask) and single global read of each graph's positions.

typedef __attribute__((ext_vector_type(2))) float v2f;
typedef __attribute__((ext_vector_type(8))) float v8f;

#define MM 1024
#define KK 32
#define TILE 16
#define LDSTRIDE 1028   // 1024 + 4: row-half (8 rows) offset = 8224 % 64 = 32 banks -> no conflicts

__global__ __launch_bounds__(512) void knn_edges_kernel(
    const float* __restrict__ pos, float* __restrict__ out, int B)
{
    __shared__ float4 p4_s[MM];              // 16 KB: x, y, z, |p|^2
    __shared__ float  d2_s[TILE * LDSTRIDE]; // ~64.3 KB (WGP has 320 KB LDS)

    const int b       = blockIdx.y;
    const int rowBase = blockIdx.x * TILE;
    const int tid     = threadIdx.x;
    const int lane    = tid & 31;
    const int w       = tid >> 5;            // wave id, 0..15

    const float* gpos = pos + (size_t)b * MM * 3;

    // ---- Phase 1: stage coordinates + squared norms into LDS ----
    for (int n = tid; n < MM; n += 512) {
        float x = gpos[n * 3 + 0];
        float y = gpos[n * 3 + 1];
        float z = gpos[n * 3 + 2];
        float4 p; p.x = x; p.y = y; p.z = z; p.w = x * x + y * y + z * z;
        p4_s[n] = p;
    }
    __syncthreads();

    // ---- Phase 2: Gram tiles via V_WMMA_F32_16X16X4_F32, 4 column-tiles per wave ----
    // 16x4 f32 A layout: lane%16 = M row; lane/16 selects K pair {0,1} / {2,3};
    // VGPR0 holds K=kb, VGPR1 holds K=kb+1. K=3 is zero padding (only 3 coords).
    const int  half = lane >> 4;
    const int  l16  = lane & 15;
    const bool hi   = (half != 0);           // K pair {2,3}

    const float4 pr4 = p4_s[rowBase + l16];
    v2f a;
    a.x = hi ? pr4.z : pr4.x;                // K = 2 or 0
    a.y = hi ? 0.0f  : pr4.y;                // K = 3 (pad) or 1

    // row squared norms for this tile's 8 output rows (tile-invariant)
    float sqr[8];
    #pragma unroll
    for (int r = 0; r < 8; ++r)
        sqr[r] = p4_s[rowBase + r + half * 8].w;

    for (int t = w * 4; t < w * 4 + 4; ++t) {
        const int colBase = t * TILE;
        const float4 pc4 = p4_s[colBase + l16];
        v2f bm;
        bm.x = hi ? pc4.z : pc4.x;
        bm.y = hi ? 0.0f  : pc4.y;
        const float sqc = pc4.w;

        v8f acc = {};
        // (neg_a, A, neg_b, B, c_mod, C, reuse_a, reuse_b)
        acc = __builtin_amdgcn_wmma_f32_16x16x4_f32(
            false, a, false, bm, (short)0, acc, false, false);

        // D layout: VGPR r, lanes 0-15 -> M=r, lanes 16-31 -> M=r+8; N = lane%16
        #pragma unroll
        for (int r = 0; r < 8; ++r) {
            const int ml = r + half * 8;
            const int nl = l16;
            float d2 = sqr[r] + sqc - 2.0f * acc[r];
            d2 = fmaxf(d2, 0.0f);                                  // numerical floor
            d2 = (rowBase + ml == colBase + nl) ? __builtin_inff() // self-edge
                                                : d2;
            d2_s[ml * LDSTRIDE + colBase + nl] = d2;
        }
    }
    __syncthreads();

    // ---- Phase 3: top-32 smallest per row; wave w owns row w ----
    // Lane holds candidates col = lane + 32*j (conflict-free LDS reads).
    float v[32];
    #pragma unroll
    for (int j = 0; j < 32; ++j)
        v[j] = d2_s[w * LDSTRIDE + (lane + 32 * j)];

    float        myD2  = __builtin_inff();
    unsigned int myCol = 0;
    for (int i = 0; i < KK; ++i) {
        // local min with index (unrolled cmp/cndmask chain)
        float bv = v[0];
        int   bj = 0;
        #pragma unroll
        for (int j = 1; j < 32; ++j)
            if (v[j] < bv) { bv = v[j]; bj = j; }

        // pack (float bits || column) -> lexicographic min == (min d2, min idx),
        // valid since d2 >= 0 so f32 bit pattern is monotone as uint.
        unsigned long long key =
            ((unsigned long long)__float_as_uint(bv) << 32) |
            (unsigned int)(bj * 32 + lane);
        #pragma unroll
        for (int off = 16; off >= 1; off >>= 1) {
            unsigned long long o = __shfl_xor(key, off, 32);
            key = (o < key) ? o : key;
        }

        const unsigned int wcol = (unsigned int)key;
        if ((wcol & 31u) == (unsigned int)lane) {
            const int wj = (int)(wcol >> 5);
            #pragma unroll
            for (int j = 0; j < 32; ++j)
                if (j == wj) v[j] = __builtin_inff();   // remove winner
        }
        if (lane == i) {                                // lane i keeps rank i
            myD2  = __uint_as_float((unsigned int)(key >> 32));
            myCol = wcol;
        }
    }

    // ---- Phase 4: outputs, tuple flattened in return order as f32 ----
    // [0, 2*BMK): edge_index (src plane then dst plane)
    // [2*BMK, 3*BMK): dist
    // [3*BMK, 4*BMK): valid (1.0/0.0)
    const int    m    = rowBase + w;
    const size_t BMK  = (size_t)B * MM * KK;
    const size_t e    = ((size_t)b * MM + m) * KK + lane;
    const float  dist = sqrtf(myD2);
    const bool   val  = dist <= 5.0f;
    const int    dst  = b * MM + m;
    const int    src  = val ? (int)((unsigned)(b * MM) + myCol) : -1;

    out[e]           = (float)src;
    out[BMK + e]     = (float)dst;
    out[2 * BMK + e] = dist;
    out[3 * BMK + e] = val ? 1.0f : 0.0f;
}

extern "C" void kernel_launch(void* const* d_in, const int* in_sizes, int n_in,
                              void* d_out, int out_size, void* d_ws, size_t ws_size,
                              hipStream_t stream) {
    const float* pos = (const float*)d_in[0];
    const int B = in_sizes[0] / (MM * 3);   // [B, M, 3] f32
    dim3 grid(MM / TILE, B);
    knn_edges_kernel<<<grid, 512, 0, stream>>>(pos, (float*)d_out, B);
}